// ExLRestSelfAtten_63436666962665
// MI455X (gfx1250) — compile-verified
//
#include <hip/hip_runtime.h>

// ---------------------------------------------------------------------------
// ExLRestSelfAtten, MI455X (gfx1250) fused single-kernel implementation.
//
// Reference collapses: softmax over a singleton axis -> atten weights are all
// ones (W_q, W_k dead).  ctx = (window-sum of h) @ W_v, and
// out = ctx @ W_out = hsum @ (W_v @ W_out)  (tiny [256,2] fused matrix).
// Only real GEMM left: h = relu(x @ W_in + b_in), done with
// V_WMMA_F32_16X16X4_F32 (full fp32 precision, wave32 WMMA).
// x is read from HBM exactly once; h lives only in LDS.
// ---------------------------------------------------------------------------

typedef __attribute__((ext_vector_type(2))) float v2f;
typedef __attribute__((ext_vector_type(4))) float v4f;
typedef __attribute__((ext_vector_type(8))) float v8f;

#define B_      16
#define S_      2048
#define D_      256     // D_IN == H == 256
#define DOUT_   2
#define PAD_    2       // ATTEN_SIZE
#define WIN_    5
#define STILE   128     // s-rows of output per workgroup
#define HROWS   132     // STILE + 2*PAD_ (halo rows of h needed)
#define MT      9       // ceil(HROWS/16) WMMA M-tiles -> 144 padded rows
#define MROWS   144
#define LDSTR   260     // LDS row stride in floats (bank-conflict-free, 16B aligned)
#define NWAVES  16
#define NTHREADS 512

static constexpr int SMEM_X      = MROWS * LDSTR;            // x tile, reused as h tile
static constexpr int SMEM_FLOATS = SMEM_X + 2 * D_ + D_;     // + Wvo[256][2] + b_in[256]

__global__ __launch_bounds__(NTHREADS, 1)
void exl_fused_kernel(const float* __restrict__ x,
                      const float* __restrict__ Win,
                      const float* __restrict__ bin,
                      const float* __restrict__ Wv,
                      const float* __restrict__ Wout,
                      const float* __restrict__ bout,
                      float* __restrict__ out)
{
    extern __shared__ float smem[];
    float* sX   = smem;                 // [MROWS][LDSTR]: x tile, later h tile
    float* sWvo = smem + SMEM_X;        // [256][2] fused W_v @ W_out
    float* sBin = sWvo + 2 * D_;        // [256] b_in

    const int tid = threadIdx.x;
    const int b   = blockIdx.x >> 4;    // S_/STILE == 16 tiles per batch
    const int s0  = (blockIdx.x & 15) * STILE;

    // ---------------- Phase A: stage x tile (zero-padded) + bias + Wvo -----
    for (int idx = tid; idx < MROWS * (D_ / 4); idx += NTHREADS) {
        const int r   = idx >> 6;          // 0..143
        const int c4  = idx & 63;
        const int srow = s0 - PAD_ + r;    // sequence row for this LDS row
        v4f v = {0.f, 0.f, 0.f, 0.f};
        if (r < HROWS && srow >= 0 && srow < S_) {
            v = *(const v4f*)(x + (((b * S_ + srow) << 8) + (c4 << 2)));
        }
        *(v4f*)(sX + r * LDSTR + (c4 << 2)) = v;
    }
    if (tid < D_) sBin[tid] = bin[tid];
    if (tid < 2 * D_) {                    // Wvo[k][o] = sum_j W_v[k][j]*W_out[j][o]
        const int k = tid >> 1, o = tid & 1;
        const float* wv = Wv + k * D_;
        float acc = 0.f;
        for (int j = 0; j < D_; ++j) acc += wv[j] * Wout[j * DOUT_ + o];
        sWvo[tid] = acc;
    }
    __syncthreads();

    // ---------------- Phase B: h = relu(x @ W_in + b_in) via f32 WMMA ------
    // wave w owns N-tile w (cols n0..n0+15); 9 M-tile accumulators live.
    const int wave  = tid >> 5;
    const int lane  = tid & 31;
    const int lhalf = lane >> 4;           // 0: K={0,1}/rows 0-7, 1: K={2,3}/rows 8-15
    const int l15   = lane & 15;
    const int n0    = wave << 4;

    v8f acc[MT];
#pragma unroll
    for (int m = 0; m < MT; ++m) {
        v8f z = {0.f, 0.f, 0.f, 0.f, 0.f, 0.f, 0.f, 0.f};
        acc[m] = z;
    }

    const float* wb = Win + (2 * lhalf) * D_ + n0 + l15;   // B frag base
#pragma unroll 1
    for (int k = 0; k < D_ / 4; ++k) {
        // B frag (4x16): VGPR0 = K row (4k + 2*lhalf), VGPR1 = next K row
        const float* wp = wb + (k << 2) * D_;
        v2f bf;
        bf.x = wp[0];
        bf.y = wp[D_];
        // A frags (16x4): lane l15 = M row, lhalf picks K pair {0,1}/{2,3}
        const float* ap = sX + l15 * LDSTR + (k << 2) + (lhalf << 1);
#pragma unroll
        for (int m = 0; m < MT; ++m) {
            v2f af = *(const v2f*)(ap + m * 16 * LDSTR);
            acc[m] = __builtin_amdgcn_wmma_f32_16x16x4_f32(
                false, af, false, bf, (short)0, acc[m], false, false);
        }
    }
    __syncthreads();   // all waves done reading x tile before overwrite

    // Store relu(h) over the x region; force exact zeros on out-of-batch halo
    {
        const float bias = sBin[n0 + l15];
#pragma unroll
        for (int m = 0; m < MT; ++m) {
            const int rbase = m * 16 + 8 * lhalf;
            float* hp = sX + rbase * LDSTR + n0 + l15;
#pragma unroll
            for (int v = 0; v < 8; ++v) {
                const int srow = s0 - PAD_ + rbase + v;
                float val = acc[m][v] + bias;
                val = val > 0.f ? val : 0.f;
                if (srow < 0 || srow >= S_) val = 0.f;   // zero padding of h
                hp[v * LDSTR] = val;
            }
        }
    }
    __syncthreads();

    // ---------------- Phase C: window-sum + tiny GEMV + outputs ------------
    const float b0 = bout[0];
    const float b1 = bout[1];
#pragma unroll 1
    for (int rr = 0; rr < STILE / NWAVES; ++rr) {
        const int r = (wave << 3) + rr;          // 0..127 -> s = s0 + r
        float p0 = 0.f, p1 = 0.f;
        for (int c = lane; c < D_; c += 32) {    // conflict-free: lanes on consecutive cols
            const float* hp = sX + r * LDSTR + c;
            const float hs = hp[0] + hp[LDSTR] + hp[2 * LDSTR]
                           + hp[3 * LDSTR] + hp[4 * LDSTR];
            p0 += hs * sWvo[2 * c + 0];
            p1 += hs * sWvo[2 * c + 1];
        }
#pragma unroll
        for (int off = 16; off > 0; off >>= 1) {
            p0 += __shfl_xor(p0, off, 32);
            p1 += __shfl_xor(p1, off, 32);
        }
        if (lane == 0) {
            float* op = out + (b * S_ + s0 + r) * DOUT_;
            op[0] = p0 + b0;
            op[1] = p1 + b1;
        }
    }

    // atten_weights: softmax over singleton axis == all ones [B,S,1,5]
    float* att = out + B_ * S_ * DOUT_ + (b * S_ + s0) * WIN_;
    for (int idx = tid; idx < STILE * WIN_; idx += NTHREADS) att[idx] = 1.0f;
}

extern "C" void kernel_launch(void* const* d_in, const int* in_sizes, int n_in,
                              void* d_out, int out_size, void* d_ws, size_t ws_size,
                              hipStream_t stream) {
    const float* x    = (const float*)d_in[0];
    const float* Win  = (const float*)d_in[1];
    const float* bin  = (const float*)d_in[2];
    // d_in[3] = W_q, d_in[4] = W_k : dead (softmax over singleton axis)
    const float* Wv   = (const float*)d_in[5];
    const float* Wout = (const float*)d_in[6];
    const float* bout = (const float*)d_in[7];
    float* out = (float*)d_out;

    const dim3 grid(B_ * (S_ / STILE));          // 256 workgroups
    const size_t smem = SMEM_FLOATS * sizeof(float);   // ~150 KB of the 320 KB WGP LDS
    exl_fused_kernel<<<grid, NTHREADS, smem, stream>>>(x, Win, bin, Wv, Wout, bout, out);
}